// SPDNonLinearity_61117384622481
// MI455X (gfx1250) — compile-verified
//
#include <hip/hip_runtime.h>
#include <hip/hip_bf16.h>

typedef __attribute__((ext_vector_type(2))) float v2f;
typedef __attribute__((ext_vector_type(8))) float v8f;

#define WPB      4      // waves (matrices) per block; 128 threads
#define NSWEEPS  10     // cyclic-Jacobi sweeps (32x32 converges in ~8)
#define LDC      33     // padded LDS column stride (odd -> conflict-free on 64 banks)
#define DIM      32

// Wave-local LDS sync: each wave owns its matrix exclusively, so instead of a
// block-wide s_barrier we only need (a) a compiler fence so ds ops are not
// reordered across the point, and (b) a DScnt drain so the wave's own LDS
// stores are complete before its subsequent cross-lane LDS loads.
__device__ __forceinline__ void wave_lds_sync() {
  asm volatile("s_wait_dscnt 0x0" ::: "memory");
}

__global__ __launch_bounds__(WPB * 32)
void SPDNonLinearity_61117384622481_kernel(const float* __restrict__ in,
                                           float* __restrict__ out,
                                           int nbatch) {
  __shared__ float Ash[WPB][LDC * DIM];
  __shared__ float Vsh[WPB][LDC * DIM];
  __shared__ float Csh[WPB][DIM];
  __shared__ float Ssh[WPB][DIM];
  __shared__ float Esh[WPB][DIM];

  const int lane = threadIdx.x & 31;
  const int wid  = threadIdx.x >> 5;
  int batch = blockIdx.x * WPB + wid;
  if (batch >= nbatch) batch = nbatch - 1;   // redundant recompute, keeps EXEC full

  float* __restrict__ A  = Ash[wid];
  float* __restrict__ V  = Vsh[wid];
  float* __restrict__ Cc = Csh[wid];
  float* __restrict__ Sc = Ssh[wid];
  float* __restrict__ Ew = Esh[wid];

  const float* __restrict__ src = in + (size_t)batch * (DIM * DIM);

  // Load A[row][col] -> A[col*LDC+row]; at fixed i the 32 lanes read 128
  // consecutive bytes (coalesced).
  #pragma unroll 4
  for (int i = 0; i < DIM; ++i) A[lane * LDC + i] = src[i * DIM + lane];
  // V = I
  #pragma unroll 4
  for (int i = 0; i < DIM; ++i) V[lane * LDC + i] = (i == lane) ? 1.0f : 0.0f;
  wave_lds_sync();
  // Symmetrize from the UPPER triangle (UPLO='U'): lower[i][l] (i>l) = upper A[l][i].
  for (int i = lane + 1; i < DIM; ++i) A[lane * LDC + i] = A[i * LDC + lane];
  wave_lds_sync();

  // ---- Cyclic Jacobi, XOR tournament schedule: 16 disjoint rotations/round ----
  for (int sweep = 0; sweep < NSWEEPS; ++sweep) {
    for (int r = 1; r < DIM; ++r) {
      const int m = lane ^ r;
      const int p = lane < m ? lane : m;
      const int q = lane < m ? m : lane;

      const float app = A[p * LDC + p];
      const float aqq = A[q * LDC + q];
      const float apq = A[q * LDC + p];

      float c = 1.0f, s = 0.0f;
      if (fabsf(apq) > 1e-12f) {
        float tau = (aqq - app) / (2.0f * apq);
        float t   = copysignf(1.0f, tau) / (fabsf(tau) + sqrtf(1.0f + tau * tau));
        c = rsqrtf(1.0f + t * t);
        s = t * c;
      }
      Cc[lane] = c;
      Sc[lane] = s;
      const float ss = (lane < m) ? -s : s;   // col'_p = c*p - s*q ; col'_q = s*p + c*q

      // A <- A*J, V <- V*J : lane owns column `lane`, partner column `m`.
      float na[DIM], nv[DIM];
      #pragma unroll
      for (int i = 0; i < DIM; ++i) {
        na[i] = c * A[lane * LDC + i] + ss * A[m * LDC + i];
        nv[i] = c * V[lane * LDC + i] + ss * V[m * LDC + i];
      }
      wave_lds_sync();                        // all cross-column reads done
      #pragma unroll
      for (int i = 0; i < DIM; ++i) {
        A[lane * LDC + i] = na[i];
        V[lane * LDC + i] = nv[i];
      }
      wave_lds_sync();                        // columns + (c,s) visible

      // A <- J^T * A : lane-local row-pair fixups inside its own column.
      for (int j = 0; j < DIM; ++j) {
        const int k = j ^ r;
        if (j < k) {
          const float cj = Cc[j], sj = Sc[j];
          const float x = A[lane * LDC + j];
          const float y = A[lane * LDC + k];
          A[lane * LDC + j] = cj * x - sj * y;
          A[lane * LDC + k] = sj * x + cj * y;
        }
      }
      wave_lds_sync();                        // ready for next round's (c,s)
    }
  }

  // ---- Eigenvalue nonlinearity: rank among ascending order, clamp, +0.1*rank ----
  const float e = A[lane * LDC + lane];
  int rank = 0;
  #pragma unroll
  for (int j = 0; j < DIM; ++j) {
    const float ej = __shfl(e, j, 32);
    rank += ((ej < e) || (ej == e && j < lane)) ? 1 : 0;
  }
  Ew[lane] = fmaxf(e, 0.5f) + 0.1f * (float)rank;
  wave_lds_sync();

  // ---- out = (V*diag(ew)) * V^T via V_WMMA_F32_16X16X4_F32 (4 tiles x 8 steps) ----
  float* __restrict__ dst = out + (size_t)batch * (DIM * DIM);
  const int half = lane >> 4;     // K-half selector for A/B operands
  const int t0   = lane & 15;     // M (for A) / N (for B) within the 16-wide tile

  #pragma unroll
  for (int I = 0; I < DIM; I += 16) {
    #pragma unroll
    for (int J = 0; J < DIM; J += 16) {
      v8f acc = {};
      #pragma unroll
      for (int kk = 0; kk < DIM; kk += 4) {
        const int kb = kk + 2 * half;
        v2f a, b;
        // A = W[I..I+15, kb..kb+1], W[m][k] = V[m][k]*ew[k]; V[m][k] at V[k*LDC+m]
        a.x = V[(kb + 0) * LDC + (I + t0)] * Ew[kb + 0];
        a.y = V[(kb + 1) * LDC + (I + t0)] * Ew[kb + 1];
        // B = V^T[kb..kb+1, J..J+15] = V[J+t0][k]
        b.x = V[(kb + 0) * LDC + (J + t0)];
        b.y = V[(kb + 1) * LDC + (J + t0)];
        acc = __builtin_amdgcn_wmma_f32_16x16x4_f32(
            /*neg_a=*/false, a, /*neg_b=*/false, b,
            /*c_mod=*/(short)0, acc, /*reuse_a=*/false, /*reuse_b=*/false);
      }
      // C/D layout: VGPR v, lanes 0-15 -> M=v, lanes 16-31 -> M=v+8; N=lane&15
      #pragma unroll
      for (int vv = 0; vv < 8; ++vv) {
        dst[(I + vv + 8 * half) * DIM + (J + t0)] = acc[vv];
      }
    }
  }
}

extern "C" void kernel_launch(void* const* d_in, const int* in_sizes, int n_in,
                              void* d_out, int out_size, void* d_ws, size_t ws_size,
                              hipStream_t stream) {
  const float* in = (const float*)d_in[0];
  float* out = (float*)d_out;
  const int nbatch = in_sizes[0] / (DIM * DIM);   // 50000
  const int blocks = (nbatch + WPB - 1) / WPB;    // 12500
  SPDNonLinearity_61117384622481_kernel<<<blocks, WPB * 32, 0, stream>>>(in, out, nbatch);
}